// SimpleGaussianModel_43138651521251
// MI455X (gfx1250) — compile-verified
//
#include <hip/hip_runtime.h>
#include <hip/hip_bf16.h>

typedef float v2f __attribute__((ext_vector_type(2)));
typedef float v8f __attribute__((ext_vector_type(8)));

#define NPTS   8192
#define NS     32
#define NB     1024
#define NQ     (NB * NS)        // 32768 query points
#define TILE_Q 16
#define NTILES (NQ / TILE_Q)    // 2048 waves
#define WAVES_PER_BLOCK 8

// ---------------------------------------------------------------------------
// Prep: per-point precompute sc[n] = { sig(op)*sig(c0..c2), |xyz|^2 } into ws,
// zero rgb output, write constant depth (mean of linspace(2,4,32) == 3.0).
// ---------------------------------------------------------------------------
__global__ __launch_bounds__(256) void prep_kernel(
    const float* __restrict__ xyz, const float* __restrict__ colors,
    const float* __restrict__ opacity, float* __restrict__ sc,
    float* __restrict__ out, int use_sc) {
  int i = blockIdx.x * blockDim.x + threadIdx.x;
  if (i < NPTS && use_sc) {
    float x = xyz[3 * i + 0], y = xyz[3 * i + 1], z = xyz[3 * i + 2];
    float xsq = x * x + y * y + z * z;
    float so = 1.0f / (1.0f + __expf(-opacity[i]));
    float s0 = so / (1.0f + __expf(-colors[3 * i + 0]));
    float s1 = so / (1.0f + __expf(-colors[3 * i + 1]));
    float s2 = so / (1.0f + __expf(-colors[3 * i + 2]));
    float4 v = make_float4(s0, s1, s2, xsq);
    *(float4*)(sc + 4 * i) = v;
  }
  if (i < 3 * NB) out[i] = 0.0f;          // rgb accumulators
  if (i < NB) out[3 * NB + i] = 3.0f;     // depth = mean(linspace(2,4,32))
}

// ---------------------------------------------------------------------------
// Main: one wave per 16-query tile; V_WMMA_F32_16X16X4_F32 computes a
// 16x16 tile of p.x dot products per iteration (K=3 padded to 4).
// ---------------------------------------------------------------------------
__global__ __launch_bounds__(256) void splat_kernel(
    const float* __restrict__ rays_o, const float* __restrict__ rays_d,
    const float* __restrict__ xyz, const float* __restrict__ sc,
    const float* __restrict__ colors, const float* __restrict__ opacity,
    float* __restrict__ out, int use_sc) {
  const int lane = threadIdx.x & 31;
  const int wave = threadIdx.x >> 5;
  const int tile = blockIdx.x * WAVES_PER_BLOCK + wave;   // 0..NTILES-1 exact

  const int  li = lane & 15;
  const bool hiHalf = lane >= 16;

  // ---- query point for this lane's row of the A matrix -----------------
  const int q = tile * TILE_Q + li;        // 16 consecutive q, all same ray b
  const int b = q >> 5;
  const int s = q & 31;
  const float t = 2.0f + (float)s * (2.0f / 31.0f);
  const float px = rays_o[3 * b + 0] + rays_d[3 * b + 0] * t;
  const float py = rays_o[3 * b + 1] + rays_d[3 * b + 1] * t;
  const float pz = rays_o[3 * b + 2] + rays_d[3 * b + 2] * t;
  const float psq = px * px + py * py + pz * pz;

  // A matrix (16x4 f32): lanes 0-15 hold {K0,K1}={x,y}, lanes 16-31 {K2,K3}={z,0}
  v2f A;
  A.x = hiHalf ? pz : px;
  A.y = hiHalf ? 0.0f : py;

  // |p|^2 for rows M = r + 8*hiHalf, broadcast from the lane that owns query M
  float psq8[8];
#pragma unroll
  for (int r = 0; r < 8; ++r)
    psq8[r] = __shfl(psq, r + (hiHalf ? 8 : 0), 32);

  const float kExp = -5.0f * 1.4426950408889634f;  // exp(-5d) = exp2(kExp*d)
  float acc0 = 0.0f, acc1 = 0.0f, acc2 = 0.0f;

  for (int n0 = 0; n0 < NPTS; n0 += 16) {
    const int n = n0 + li;
    const float xx = xyz[3 * n + 0];
    const float xy = xyz[3 * n + 1];
    const float xz = xyz[3 * n + 2];

    // B matrix (4x16 f32): lanes 0-15 hold {K0,K1}={x,y}, lanes 16-31 {K2,K3}={z,0}
    v2f Bm;
    Bm.x = hiHalf ? xz : xx;
    Bm.y = hiHalf ? 0.0f : xy;

    float s0, s1, s2, xsq;
    if (use_sc) {
      const float4 v = *(const float4*)(sc + 4 * n);
      s0 = v.x; s1 = v.y; s2 = v.z; xsq = v.w;
    } else {
      xsq = xx * xx + xy * xy + xz * xz;
      const float so = 1.0f / (1.0f + __expf(-opacity[n]));
      s0 = so / (1.0f + __expf(-colors[3 * n + 0]));
      s1 = so / (1.0f + __expf(-colors[3 * n + 1]));
      s2 = so / (1.0f + __expf(-colors[3 * n + 2]));
    }

    // cross[M][N] tile: M = r + 8*hiHalf, N = n0 + li
    v8f c = {};
    c = __builtin_amdgcn_wmma_f32_16x16x4_f32(
        /*neg_a=*/false, A, /*neg_b=*/false, Bm,
        /*c_mod=*/(short)0, c, /*reuse_a=*/false, /*reuse_b=*/false);

    float wsum = 0.0f;
#pragma unroll
    for (int r = 0; r < 8; ++r) {
      float d2 = psq8[r] + xsq - 2.0f * c[r];
      float dist = __builtin_amdgcn_sqrtf(__builtin_fmaxf(d2, 0.0f));
      wsum += __builtin_amdgcn_exp2f(kExp * dist);
    }
    // rgb partial: color depends only on n (this lane), so factor out wsum
    acc0 = __builtin_fmaf(wsum, s0, acc0);
    acc1 = __builtin_fmaf(wsum, s1, acc1);
    acc2 = __builtin_fmaf(wsum, s2, acc2);
  }

  // wave32 reduction: sum over all lanes (covers all 16 rows x all n columns)
#pragma unroll
  for (int off = 16; off >= 1; off >>= 1) {
    acc0 += __shfl_xor(acc0, off, 32);
    acc1 += __shfl_xor(acc1, off, 32);
    acc2 += __shfl_xor(acc2, off, 32);
  }

  if (lane == 0) {
    const int bb = tile >> 1;              // two tiles (32 samples) per ray
    const float invS = 1.0f / (float)NS;
    atomicAdd(&out[3 * bb + 0], acc0 * invS);
    atomicAdd(&out[3 * bb + 1], acc1 * invS);
    atomicAdd(&out[3 * bb + 2], acc2 * invS);
  }
}

extern "C" void kernel_launch(void* const* d_in, const int* in_sizes, int n_in,
                              void* d_out, int out_size, void* d_ws, size_t ws_size,
                              hipStream_t stream) {
  const float* rays_o  = (const float*)d_in[0];
  const float* rays_d  = (const float*)d_in[1];
  const float* xyz     = (const float*)d_in[2];
  const float* colors  = (const float*)d_in[3];
  const float* opacity = (const float*)d_in[4];
  float* out = (float*)d_out;
  float* sc  = (float*)d_ws;

  const int use_sc = (ws_size >= (size_t)NPTS * 4 * sizeof(float)) ? 1 : 0;

  prep_kernel<<<(NPTS + 255) / 256, 256, 0, stream>>>(
      xyz, colors, opacity, sc, out, use_sc);

  splat_kernel<<<NTILES / WAVES_PER_BLOCK, 256, 0, stream>>>(
      rays_o, rays_d, xyz, sc, colors, opacity, out, use_sc);
}